// NTMCell_15006615732960
// MI455X (gfx1250) — compile-verified
//
#include <hip/hip_runtime.h>
#include <hip/hip_bf16.h>

typedef __attribute__((ext_vector_type(2))) float v2f;
typedef __attribute__((ext_vector_type(8))) float v8f;

#define NN   8192   // memory rows
#define WW   1024   // word size
#define CC   2048   // controller size (H)
#define DIN  4096
#define DOUT 2048
#define DL   5120   // D_IN + W
#define FH   8192   // 4*H

__device__ __forceinline__ float sigmoidf_dev(float x) {
    return 1.0f / (1.0f + expf(-x));
}
__device__ __forceinline__ float softplusf_dev(float x) {
    return (x > 0.0f) ? (x + log1pf(expf(-x))) : log1pf(expf(x));
}

__device__ __forceinline__ v2f ld2(const float* p) {
    const float2 t = *reinterpret_cast<const float2*>(p);
    v2f r; r[0] = t.x; r[1] = t.y;
    return r;
}

// ---------------------------------------------------------------------------
// y[m] = dot(A[m, 0:K], x[0:K])   (A row-major, leading dim lda)
// grid.x = M/16, block = 128 (4 waves). Each wave owns K/4; requires
// (K/4) % 16 == 0 (true for K = 1024, 2048, 5120).
// Uses V_WMMA_F32_16X16X4_F32 with the vector broadcast across all 16 B
// columns, so every column of the D tile carries the same 16 dot products.
// K loop unrolled 4x so 8 loads are in flight per WMMA group (avoids the
// compiler's full s_wait_loadcnt 0 drain before every WMMA).
// ---------------------------------------------------------------------------
__global__ __launch_bounds__(128) void gemv16_wmma(
    const float* __restrict__ A, const float* __restrict__ x,
    float* __restrict__ y, int K, int lda) {
    __shared__ float red[64];                 // 4 waves * 16 rows
    const int tid  = threadIdx.x;
    const int lane = tid & 31;
    const int wave = tid >> 5;
    const int row0 = blockIdx.x * 16;
    const int m    = lane & 15;
    const int kh   = (lane >> 4) << 1;        // 0 for lanes 0-15, 2 for lanes 16-31
    const int kPerWave = K >> 2;
    const int kBegin   = wave * kPerWave;
    const int kEnd     = kBegin + kPerWave;

    const float* __restrict__ arow = A + (size_t)(row0 + m) * (size_t)lda + kh;
    const float* __restrict__ xv   = x + kh;

    v8f acc = {};
    for (int k = kBegin; k < kEnd; k += 16) {
        const v2f a0 = ld2(arow + k);
        const v2f a1 = ld2(arow + k + 4);
        const v2f a2 = ld2(arow + k + 8);
        const v2f a3 = ld2(arow + k + 12);
        const v2f b0 = ld2(xv + k);
        const v2f b1 = ld2(xv + k + 4);
        const v2f b2 = ld2(xv + k + 8);
        const v2f b3 = ld2(xv + k + 12);
        acc = __builtin_amdgcn_wmma_f32_16x16x4_f32(false, a0, false, b0, (short)0, acc, false, false);
        acc = __builtin_amdgcn_wmma_f32_16x16x4_f32(false, a1, false, b1, (short)0, acc, false, false);
        acc = __builtin_amdgcn_wmma_f32_16x16x4_f32(false, a2, false, b2, (short)0, acc, false, false);
        acc = __builtin_amdgcn_wmma_f32_16x16x4_f32(false, a3, false, b3, (short)0, acc, false, false);
    }
    // D layout: VGPR r -> row r (lanes 0-15) / row r+8 (lanes 16-31); all N equal.
    if (lane == 0 || lane == 16) {
        const int base = wave * 16 + ((lane == 16) ? 8 : 0);
#pragma unroll
        for (int r = 0; r < 8; ++r) red[base + r] = acc[r];
    }
    __syncthreads();
    if (tid < 16) {
        y[row0 + tid] = red[tid] + red[16 + tid] + red[32 + tid] + red[48 + tid];
    }
}

// ---------------------------------------------------------------------------
// Deterministic column reduction:  p1[c][j] = sum_{n in chunk c} v1[n]*M[n][j]
// (optionally p2 with v2). grid = (Wcols/256, nChunks), block = 256.
// ---------------------------------------------------------------------------
__global__ void colsum_partial(const float* __restrict__ mem,
                               const float* __restrict__ v1,
                               const float* __restrict__ v2,
                               float* __restrict__ p1, float* __restrict__ p2,
                               int N, int Wc, int nChunks) {
    const int j  = blockIdx.x * blockDim.x + threadIdx.x;
    const int c  = blockIdx.y;
    const int per = N / nChunks;
    const int n0 = c * per, n1 = n0 + per;
    float a1 = 0.f, a2 = 0.f;
    for (int n = n0; n < n1; ++n) {
        const float mv = mem[(size_t)n * Wc + j];
        a1 += v1[n] * mv;
        if (v2) a2 += v2[n] * mv;
    }
    p1[(size_t)c * Wc + j] = a1;
    if (p2) p2[(size_t)c * Wc + j] = a2;
}

__global__ void colsum_reduce(const float* __restrict__ p1, const float* __restrict__ p2,
                              float* __restrict__ o1, float* __restrict__ o2,
                              int Wc, int nChunks) {
    const int j = blockIdx.x * blockDim.x + threadIdx.x;
    if (j >= Wc) return;
    float s1 = 0.f, s2 = 0.f;
    for (int c = 0; c < nChunks; ++c) {
        s1 += p1[(size_t)c * Wc + j];
        if (p2) s2 += p2[(size_t)c * Wc + j];
    }
    o1[j] = s1;
    if (o2) o2[j] = s2;
}

// ---------------------------------------------------------------------------
// softmax over n of  scale * val[i], where scale = softplus'ed beta * invW and
//   val[i] = t1[i]                              if t2 == nullptr
//   val[i] = t1[i] + w2[i]*(s - t2[i])          otherwise  (key3 . mem2[n])
// Single block of 1024 threads.
// ---------------------------------------------------------------------------
__global__ __launch_bounds__(1024) void softmax_k(
    const float* __restrict__ t1, const float* __restrict__ t2,
    const float* __restrict__ w2v, const float* __restrict__ sPtr,
    const float* __restrict__ betaPtr, float* __restrict__ out,
    int n, float invW) {
    __shared__ float sdata[1024];
    const int tid = threadIdx.x;
    const int nt  = blockDim.x;
    const float scale = betaPtr[0] * invW;
    const float sv    = (sPtr != nullptr) ? sPtr[0] : 0.0f;

    float lmax = -3.0e38f;
    for (int i = tid; i < n; i += nt) {
        float v = t1[i];
        if (t2) v += w2v[i] * (sv - t2[i]);
        lmax = fmaxf(lmax, v * scale);
    }
    sdata[tid] = lmax; __syncthreads();
    for (int s = nt >> 1; s > 0; s >>= 1) {
        if (tid < s) sdata[tid] = fmaxf(sdata[tid], sdata[tid + s]);
        __syncthreads();
    }
    const float gmax = sdata[0];
    __syncthreads();

    float lsum = 0.f;
    for (int i = tid; i < n; i += nt) {
        float v = t1[i];
        if (t2) v += w2v[i] * (sv - t2[i]);
        lsum += expf(v * scale - gmax);
    }
    sdata[tid] = lsum; __syncthreads();
    for (int s = nt >> 1; s > 0; s >>= 1) {
        if (tid < s) sdata[tid] += sdata[tid + s];
        __syncthreads();
    }
    const float inv = 1.0f / sdata[0];
    __syncthreads();

    for (int i = tid; i < n; i += nt) {
        float v = t1[i];
        if (t2) v += w2v[i] * (sv - t2[i]);
        out[i] = expf(v * scale - gmax) * inv;
    }
}

// out[0] = (softplus?) ( sum a[i]*b[i] + bias )   -- one block of 256 threads
__global__ __launch_bounds__(256) void dot_scalar(
    const float* __restrict__ a, const float* __restrict__ b,
    const float* __restrict__ bias, float* __restrict__ out, int n, int softplus) {
    __shared__ float sdata[256];
    const int tid = threadIdx.x;
    float s = 0.f;
    for (int i = tid; i < n; i += 256) s += a[i] * b[i];
    sdata[tid] = s; __syncthreads();
    for (int k = 128; k > 0; k >>= 1) {
        if (tid < k) sdata[tid] += sdata[tid + k];
        __syncthreads();
    }
    if (tid == 0) {
        float v = sdata[0] + (bias ? bias[0] : 0.f);
        if (softplus) v = softplusf_dev(v);
        out[0] = v;
    }
}

// co0 == 0  =>  key1 = tanh(rk_b), beta1 = softplus(rb_b)
__global__ void init_key1(const float* __restrict__ rk_b, const float* __restrict__ rb_b,
                          float* __restrict__ key1, float* __restrict__ beta1, int Wc) {
    const int i = blockIdx.x * blockDim.x + threadIdx.x;
    if (i < Wc) key1[i] = tanhf(rk_b[i]);
    if (i == 0) beta1[0] = softplusf_dev(rb_b[0]);
}

__global__ void concat2(const float* __restrict__ a, int na,
                        const float* __restrict__ b, int nb, float* __restrict__ o) {
    const int i = blockIdx.x * blockDim.x + threadIdx.x;
    if (i < na) o[i] = a[i];
    else if (i < na + nb) o[i] = b[i - na];
}

// h0 = c0 = 0: forget gate dead.  c = sig(i)*tanh(g); h = sig(o)*tanh(c)
__global__ void lstm_h(const float* __restrict__ g, const float* __restrict__ b_ih,
                       const float* __restrict__ b_hh, float* __restrict__ h, int H) {
    const int t = blockIdx.x * blockDim.x + threadIdx.x;
    if (t >= H) return;
    const float gi = g[t]         + b_ih[t]         + b_hh[t];
    const float gg = g[2 * H + t] + b_ih[2 * H + t] + b_hh[2 * H + t];
    const float go = g[3 * H + t] + b_ih[3 * H + t] + b_hh[3 * H + t];
    const float c  = sigmoidf_dev(gi) * tanhf(gg);
    h[t] = sigmoidf_dev(go) * tanhf(c);
}

__global__ void head_act(const float* __restrict__ k2r, const float* __restrict__ wk_b,
                         const float* __restrict__ er,  const float* __restrict__ we_b,
                         const float* __restrict__ ar,  const float* __restrict__ wa_b,
                         const float* __restrict__ k3r, const float* __restrict__ rk_b,
                         float* __restrict__ key2, float* __restrict__ erase,
                         float* __restrict__ addv, float* __restrict__ key3,
                         float* __restrict__ key3e, int n) {
    const int i = blockIdx.x * blockDim.x + threadIdx.x;
    if (i >= n) return;
    const float kk2 = tanhf(k2r[i] + wk_b[i]);
    const float ee  = sigmoidf_dev(er[i] + we_b[i]);
    const float aa  = tanhf(ar[i] + wa_b[i]);
    const float kk3 = tanhf(k3r[i] + rk_b[i]);
    key2[i] = kk2; erase[i] = ee; addv[i] = aa; key3[i] = kk3; key3e[i] = kk3 * ee;
}

__global__ void mulvec(const float* __restrict__ a, const float* __restrict__ b,
                       float* __restrict__ o, int n) {
    const int i = blockIdx.x * blockDim.x + threadIdx.x;
    if (i < n) o[i] = a[i] * b[i];
}

// rv2 = u1 - erase*u2 + dp*add
__global__ void rv2_kernel(const float* __restrict__ u1, const float* __restrict__ u2,
                           const float* __restrict__ erase, const float* __restrict__ addv,
                           const float* __restrict__ dpPtr, float* __restrict__ rv2, int n) {
    const int i = blockIdx.x * blockDim.x + threadIdx.x;
    if (i < n) rv2[i] = u1[i] - erase[i] * u2[i] + dpPtr[0] * addv[i];
}

// d_out[0:n] = oraw + b_out ; d_out[n:2n] = h
__global__ void finalize_out(const float* __restrict__ oraw, const float* __restrict__ b_out,
                             const float* __restrict__ h, float* __restrict__ out, int n) {
    const int i = blockIdx.x * blockDim.x + threadIdx.x;
    if (i < n) {
        out[i]     = oraw[i] + b_out[i];
        out[n + i] = h[i];
    }
}

extern "C" void kernel_launch(void* const* d_in, const int* in_sizes, int n_in,
                              void* d_out, int out_size, void* d_ws, size_t ws_size,
                              hipStream_t stream) {
    (void)in_sizes; (void)n_in; (void)out_size; (void)ws_size;
    const float* x      = (const float*)d_in[0];
    // d_in[1] prev_weights: dead (gamma==1 kills the location path)
    const float* memory = (const float*)d_in[2];
    const float* rk_w   = (const float*)d_in[3];
    const float* rk_b   = (const float*)d_in[4];
    const float* rb_w   = (const float*)d_in[5];
    const float* rb_b   = (const float*)d_in[6];
    const float* wk_w   = (const float*)d_in[7];
    const float* wk_b   = (const float*)d_in[8];
    const float* wb_w   = (const float*)d_in[9];
    const float* wb_b   = (const float*)d_in[10];
    const float* we_w   = (const float*)d_in[11];
    const float* we_b   = (const float*)d_in[12];
    const float* wa_w   = (const float*)d_in[13];
    const float* wa_b   = (const float*)d_in[14];
    const float* W_ih   = (const float*)d_in[15];
    // d_in[16] W_hh: dead (h0 == 0)
    const float* b_ih   = (const float*)d_in[17];
    const float* b_hh   = (const float*)d_in[18];
    const float* W_out  = (const float*)d_in[19];
    const float* b_out  = (const float*)d_in[20];
    float* out = (float*)d_out;

    // ---- workspace carve-up (floats) ----
    float* w = (float*)d_ws;
    size_t off = 0;
    auto alloc = [&](size_t n) { float* p = w + off; off += n; return p; };
    float* key1  = alloc(WW);   float* beta1 = alloc(1);
    float* d1    = alloc(NN);   float* w1    = alloc(NN);
    float* part1 = alloc(64 * WW);
    float* part2 = alloc(64 * WW);
    float* rv1   = alloc(WW);   float* ci    = alloc(DL);
    float* gates = alloc(FH);   float* h     = alloc(CC);
    float* k2raw = alloc(WW);   float* eraw  = alloc(WW);
    float* araw  = alloc(WW);   float* k3raw = alloc(WW);
    float* key2  = alloc(WW);   float* erase = alloc(WW);
    float* addv  = alloc(WW);   float* key3  = alloc(WW);
    float* key3e = alloc(WW);
    float* beta2 = alloc(1);    float* beta3 = alloc(1);
    float* sdot  = alloc(1);    float* dp    = alloc(1);
    float* d2    = alloc(NN);   float* w2    = alloc(NN);
    float* t1    = alloc(NN);   float* t2    = alloc(NN);
    float* w3    = alloc(NN);   float* pvec  = alloc(NN);
    float* u1    = alloc(WW);   float* u2    = alloc(WW);
    float* rv2   = alloc(WW);   float* oraw  = alloc(DOUT);

    const float invW = 1.0f / (float)WW;

    // 1) first addressing with co0 == 0
    init_key1<<<(WW + 255) / 256, 256, 0, stream>>>(rk_b, rb_b, key1, beta1, WW);
    gemv16_wmma<<<NN / 16, 128, 0, stream>>>(memory, key1, d1, WW, WW);
    softmax_k<<<1, 1024, 0, stream>>>(d1, nullptr, nullptr, nullptr, beta1, w1, NN, invW);

    // 2) rv1 = w1 @ memory ; LSTM step
    colsum_partial<<<dim3(WW / 256, 64), 256, 0, stream>>>(memory, w1, nullptr, part1, nullptr, NN, WW, 64);
    colsum_reduce<<<(WW + 255) / 256, 256, 0, stream>>>(part1, nullptr, rv1, nullptr, WW, 64);
    concat2<<<(DL + 255) / 256, 256, 0, stream>>>(x, DIN, rv1, WW, ci);
    gemv16_wmma<<<FH / 16, 128, 0, stream>>>(W_ih, ci, gates, DL, DL);
    lstm_h<<<(CC + 255) / 256, 256, 0, stream>>>(gates, b_ih, b_hh, h, CC);

    // 3) head projections from h (all K = C = 2048)
    gemv16_wmma<<<WW / 16, 128, 0, stream>>>(wk_w, h, k2raw, CC, CC);
    gemv16_wmma<<<WW / 16, 128, 0, stream>>>(we_w, h, eraw,  CC, CC);
    gemv16_wmma<<<WW / 16, 128, 0, stream>>>(wa_w, h, araw,  CC, CC);
    gemv16_wmma<<<WW / 16, 128, 0, stream>>>(rk_w, h, k3raw, CC, CC);
    head_act<<<(WW + 255) / 256, 256, 0, stream>>>(k2raw, wk_b, eraw, we_b, araw, wa_b,
                                                   k3raw, rk_b, key2, erase, addv, key3, key3e, WW);
    dot_scalar<<<1, 256, 0, stream>>>(h, wb_w, wb_b, beta2, CC, 1);
    dot_scalar<<<1, 256, 0, stream>>>(h, rb_w, rb_b, beta3, CC, 1);

    // 4) write addressing w2 = softmax(beta2 * (memory @ key2)/W)
    gemv16_wmma<<<NN / 16, 128, 0, stream>>>(memory, key2, d2, WW, WW);
    softmax_k<<<1, 1024, 0, stream>>>(d2, nullptr, nullptr, nullptr, beta2, w2, NN, invW);

    // 5) read addressing on mem2 WITHOUT materializing mem2:
    //    key3.mem2[n] = t1[n] - w2[n]*t2[n] + w2[n]*(key3.add)
    gemv16_wmma<<<NN / 16, 128, 0, stream>>>(memory, key3,  t1, WW, WW);
    gemv16_wmma<<<NN / 16, 128, 0, stream>>>(memory, key3e, t2, WW, WW);
    dot_scalar<<<1, 256, 0, stream>>>(key3, addv, nullptr, sdot, WW, 0);
    softmax_k<<<1, 1024, 0, stream>>>(t1, t2, w2, sdot, beta3, w3, NN, invW);

    // 6) rv2 = w3 @ mem2 = u1 - erase*u2 + (w3.w2)*add
    mulvec<<<(NN + 255) / 256, 256, 0, stream>>>(w3, w2, pvec, NN);
    dot_scalar<<<1, 256, 0, stream>>>(w3, w2, nullptr, dp, NN, 0);
    colsum_partial<<<dim3(WW / 256, 64), 256, 0, stream>>>(memory, w3, pvec, part1, part2, NN, WW, 64);
    colsum_reduce<<<(WW + 255) / 256, 256, 0, stream>>>(part1, part2, u1, u2, WW, 64);
    rv2_kernel<<<(WW + 255) / 256, 256, 0, stream>>>(u1, u2, erase, addv, dp, rv2, WW);

    // 7) out = rv2 @ W_out.T + b_out ; emit (out, h)
    gemv16_wmma<<<DOUT / 16, 128, 0, stream>>>(W_out, rv2, oraw, WW, WW);
    finalize_out<<<(DOUT + 255) / 256, 256, 0, stream>>>(oraw, b_out, h, out, DOUT);
}